// FANSNeuralStateUpdate_Nonphys_120259084388
// MI455X (gfx1250) — compile-verified
//
#include <hip/hip_runtime.h>
#include <math.h>

// FANS neural state update: per-group tiny MLP (12->32->32->1), 16 groups,
// B=524288 samples. Compute-bound (~300 FLOP/byte) -> f32 WMMA path.
//
// Wave w in a 512-thread workgroup owns group g=w, keeps W1/W2/biases in
// registers (WMMA B layout), grid-strides over 16-batch tiles.

typedef __attribute__((ext_vector_type(2))) float v2f;
typedef __attribute__((ext_vector_type(8))) float v8f;

#define N_XB   16
#define N_XO   16
#define N_U    8
#define N_FEAT 32

__device__ __forceinline__ float tanh_fast(float x) {
#if __has_builtin(__builtin_amdgcn_tanhf)
    return __builtin_amdgcn_tanhf(x);   // v_tanh_f32 on gfx1250
#else
    return tanhf(x);
#endif
}

template <int OFF>
__device__ __forceinline__ float swz(float v) {
    return __int_as_float(__builtin_amdgcn_ds_swizzle(__float_as_int(v), OFF));
}

__global__ __launch_bounds__(512)
void fans_mlp_kernel(const float* __restrict__ x_b, const float* __restrict__ u,
                     const float* __restrict__ W1,  const float* __restrict__ b1,
                     const float* __restrict__ W2,  const float* __restrict__ b2,
                     const float* __restrict__ W3,  const float* __restrict__ b3,
                     const int*   __restrict__ idx, float* __restrict__ out,
                     int numTiles)
{
    // Per-wave transpose buffer: 32 feature-rows x stride 17 (conflict-free).
    __shared__ float hT[16][32 * 17];
    __shared__ float stage[256];   // 16 batch rows x 16 groups, one tile

    const int tid = threadIdx.x;
    const int g   = tid >> 5;       // wave id == group id
    const int l   = tid & 31;
    const int ln  = l & 15;         // lane within half (M or N index)
    const int h   = l >> 4;         // lane half (K-split for A/B operands)
    float* myT = hT[g];

    // ---- Hoist weights into registers in WMMA B layout ----
    // B operand element (k,n): lane = n + 16*(k>=2), vgpr = k&1.
    // For k-step s: vgpr v holds W[4s + 2h + v][ln + 16t].
    float w1r[3][2][2];
#pragma unroll
    for (int s = 0; s < 3; ++s)
#pragma unroll
        for (int t = 0; t < 2; ++t)
#pragma unroll
            for (int v = 0; v < 2; ++v)
                w1r[s][t][v] = W1[g * (12 * N_FEAT) + (4 * s + 2 * h + v) * N_FEAT + ln + 16 * t];

    float w2r[8][2][2];
#pragma unroll
    for (int s = 0; s < 8; ++s)
#pragma unroll
        for (int t = 0; t < 2; ++t)
#pragma unroll
            for (int v = 0; v < 2; ++v)
                w2r[s][t][v] = W2[g * (N_FEAT * N_FEAT) + (4 * s + 2 * h + v) * N_FEAT + ln + 16 * t];

    const float c1s[2] = { b1[g * N_FEAT + ln], b1[g * N_FEAT + ln + 16] };
    const float c2s[2] = { b2[g * N_FEAT + ln], b2[g * N_FEAT + ln + 16] };
    const float w3a = W3[g * N_FEAT + ln];
    const float w3b = W3[g * N_FEAT + 16 + ln];
    const float b3g = b3[g];
    const int xi0 = idx[g * 4 + 2 * h];
    const int xi1 = idx[g * 4 + 2 * h + 1];

    for (int tile = blockIdx.x; tile < numTiles; tile += gridDim.x) {
        const int row = tile * 16 + ln;

        // ---- A operand for layer 1: z = [x_b gather(4) | u(8)], 16x12 f32 ----
        // A element (m,k): lane = m + 16*(k>=2), vgpr = k&1.
        v2f a0, a1, a2;
        a0.x = x_b[row * N_XB + xi0];
        a0.y = x_b[row * N_XB + xi1];
        const float* urow = u + row * N_U;
        a1 = *(const v2f*)(urow + 2 * h);       // z cols 4..7  -> u[2h], u[2h+1]
        a2 = *(const v2f*)(urow + 4 + 2 * h);   // z cols 8..11 -> u[4+2h], u[5+2h]

        // ---- Layer 1: h1 = tanh(z @ W1 + b1), 3 K-steps x 2 N-tiles ----
        v8f h1acc[2];
#pragma unroll
        for (int t = 0; t < 2; ++t) {
            v8f c;
#pragma unroll
            for (int r = 0; r < 8; ++r) c[r] = c1s[t];
            v2f bb;
            bb.x = w1r[0][t][0]; bb.y = w1r[0][t][1];
            c = __builtin_amdgcn_wmma_f32_16x16x4_f32(false, a0, false, bb, (short)0, c, false, false);
            bb.x = w1r[1][t][0]; bb.y = w1r[1][t][1];
            c = __builtin_amdgcn_wmma_f32_16x16x4_f32(false, a1, false, bb, (short)0, c, false, false);
            bb.x = w1r[2][t][0]; bb.y = w1r[2][t][1];
            c = __builtin_amdgcn_wmma_f32_16x16x4_f32(false, a2, false, bb, (short)0, c, false, false);
#pragma unroll
            for (int r = 0; r < 8; ++r) c[r] = tanh_fast(c[r]);
            h1acc[t] = c;
        }

        // ---- Transpose h1 (C layout -> feature-major LDS, stride 17) ----
        // C element (m,n): lane = n + 16*(m>=8), vgpr = m&7  =>  m = r + 8h, n = ln + 16t.
#pragma unroll
        for (int t = 0; t < 2; ++t)
#pragma unroll
            for (int r = 0; r < 8; ++r)
                myT[(ln + 16 * t) * 17 + r + 8 * h] = h1acc[t][r];
        __builtin_amdgcn_wave_barrier();   // same-wave DS is in-order; compiler fence

        // ---- Read back as A operand for layer 2 (8 K-steps) ----
        v2f a2l[8];
#pragma unroll
        for (int s = 0; s < 8; ++s) {
            a2l[s].x = myT[(4 * s + 2 * h    ) * 17 + ln];
            a2l[s].y = myT[(4 * s + 2 * h + 1) * 17 + ln];
        }
        __builtin_amdgcn_wave_barrier();

        // ---- Layer 2: h2 = tanh(h1 @ W2 + b2), 8 K-steps x 2 N-tiles ----
        v8f h2acc[2];
#pragma unroll
        for (int t = 0; t < 2; ++t) {
            v8f c;
#pragma unroll
            for (int r = 0; r < 8; ++r) c[r] = c2s[t];
#pragma unroll
            for (int s = 0; s < 8; ++s) {
                v2f bb;
                bb.x = w2r[s][t][0]; bb.y = w2r[s][t][1];
                c = __builtin_amdgcn_wmma_f32_16x16x4_f32(false, a2l[s], false, bb, (short)0, c, false, false);
            }
#pragma unroll
            for (int r = 0; r < 8; ++r) c[r] = tanh_fast(c[r]);
            h2acc[t] = c;
        }

        // ---- Layer 3: dx = h2 @ W3 + b3. Features sit on lanes in C layout,
        // so reduce within each 16-lane half via ds_swizzle XOR tree. ----
        float dx[8];
#pragma unroll
        for (int r = 0; r < 8; ++r) {
            float p = h2acc[0][r] * w3a + h2acc[1][r] * w3b;
            p += swz<0x041F>(p);   // xor 1
            p += swz<0x081F>(p);   // xor 2
            p += swz<0x101F>(p);   // xor 4
            p += swz<0x201F>(p);   // xor 8
            dx[r] = p + b3g;       // lanes 0-15: dx[r]; lanes 16-31: dx[r+8]
        }

        // ---- Stage this wave's group-column of the 16x16 output tile ----
        if (ln == 0) {             // lanes 0 and 16 active
#pragma unroll
            for (int r = 0; r < 8; ++r)
                stage[(r + 8 * h) * N_XO + g] = dx[r];
        }
        __syncthreads();
        if (tid < 256)             // one coalesced 1KB store per tile
            out[tile * 256 + tid] = stage[tid];
        __syncthreads();
    }
}

extern "C" void kernel_launch(void* const* d_in, const int* in_sizes, int n_in,
                              void* d_out, int out_size, void* d_ws, size_t ws_size,
                              hipStream_t stream) {
    const float* x_b = (const float*)d_in[0];
    const float* u   = (const float*)d_in[1];
    const float* W1  = (const float*)d_in[2];
    const float* b1  = (const float*)d_in[3];
    const float* W2  = (const float*)d_in[4];
    const float* b2  = (const float*)d_in[5];
    const float* W3  = (const float*)d_in[6];
    const float* b3  = (const float*)d_in[7];
    const int*   idx = (const int*)d_in[8];
    float* out = (float*)d_out;

    const int B        = in_sizes[0] / N_XB;   // 524288
    const int numTiles = B / 16;               // 32768 16-batch tiles
    int blocks = numTiles < 2048 ? (numTiles > 0 ? numTiles : 1) : 2048;

    hipLaunchKernelGGL(fans_mlp_kernel, dim3(blocks), dim3(512), 0, stream,
                       x_b, u, W1, b1, W2, b2, W3, b3, idx, out, numTiles);
}